// NumericalRiskBudgeting_86517821215682
// MI455X (gfx1250) — compile-verified
//
#include <hip/hip_runtime.h>
#include <hip/hip_bf16.h>
#include <math.h>

// Problem constants (match reference).
#define BATCH 1024
#define N 256
#define NB 16          // Cholesky block size == WMMA tile size
#define NEWTON_ITERS 50
#define LDA 257        // padded LDS stride: bank((i*257+j)) = (i+j)%64 -> conflict-free columns

typedef float v2f  __attribute__((ext_vector_type(2)));
typedef float v8f  __attribute__((ext_vector_type(8)));
typedef unsigned int u32x4 __attribute__((ext_vector_type(4)));
typedef int   i32x4 __attribute__((ext_vector_type(4)));
typedef int   i32x8 __attribute__((ext_vector_type(8)));

// ---------------------------------------------------------------------------
// TDM: DMA one 256x256 f32 tile (row-major, contiguous) from global memory into
// LDS with a 1-DWORD pad after every 256 DWORDs -> LDS row stride 257 floats.
// D# group0: count=1 | lds_addr | global_addr[56:0] | type=2.
// D# group1: data_size=4B, pad_enable=1, pad_interval=7 (256 DW), pad_amount=0 (1 DW),
//            tensor_dim0=256, tensor_dim1=256, tile_dim0=256, tile_dim1=256,
//            tensor_dim0_stride=256, tensor_dim1_stride=65536.
// Groups 2/3 zero (2-D tensor). Issued by ONE wave; completion via TENSORcnt.
// ---------------------------------------------------------------------------
__device__ __forceinline__ void tdm_load_sigma_padded(const float* gsrc,
                                                      unsigned int lds_byte_off) {
    unsigned long long ga = (unsigned long long)(uintptr_t)gsrc;
    u32x4 g0;
    g0[0] = 1u;                                                    // count=1, user mode
    g0[1] = lds_byte_off;                                          // lds_addr
    g0[2] = (unsigned int)(ga & 0xFFFFFFFFull);                    // global_addr[31:0]
    g0[3] = (unsigned int)((ga >> 32) & 0x01FFFFFFull) | (2u << 30); // addr[56:32] | type=2
    i32x8 g1;
    g1[0] = (int)((2u << 16) | (1u << 20) | (7u << 22));  // data_size=2(4B)|pad_en|interval=7|amount=0
    g1[1] = (int)(256u << 16);                            // tensor_dim0[15:0]=256 (bits 79:48)
    g1[2] = (int)(256u << 16);                            // tensor_dim1[15:0]=256 (bits 111:80)
    g1[3] = (int)(256u << 16);                            // tile_dim0=256       (bits 127:112)
    g1[4] = 256;                                          // tile_dim1=256       (bits 143:128)
    g1[5] = 256;                                          // tensor_dim0_stride  (bits 207:160)
    g1[6] = 0;                                            // stride0 hi | stride1 lo16
    g1[7] = 1;                                            // tensor_dim1_stride = 65536
    i32x4 z4 = {0, 0, 0, 0};
#if __clang_major__ >= 23
    i32x8 z8 = {0, 0, 0, 0, 0, 0, 0, 0};
    __builtin_amdgcn_tensor_load_to_lds(g0, g1, z4, z4, z8, 0);
#else
    __builtin_amdgcn_tensor_load_to_lds(g0, g1, z4, z4, 0);
#endif
}

// ---------------------------------------------------------------------------
// Kernel 1: Sigma[b] = S[b]^T S[b] via FP32 WMMA.  Each wave computes a 32x32
// region (2x2 WMMA tiles) to double register reuse: 8 loads -> 4 WMMAs per K4.
// ---------------------------------------------------------------------------
__global__ void __launch_bounds__(256)
gram_wmma_kernel(const float* __restrict__ S, float* __restrict__ Sigma) {
    const int block  = blockIdx.x;
    const int batch  = block >> 3;                   // 8 blocks per batch
    const int wave   = threadIdx.x >> 5;             // 8 waves per block
    const int region = ((block & 7) << 3) + wave;    // 0..63 (8x8 grid of 32x32)
    const int bi0 = (region >> 3) * 32;              // region row base
    const int bj0 = (region & 7) * 32;               // region col base
    const int lane = threadIdx.x & 31;
    const int half = lane >> 4;                      // K sub-pair / row half select
    const int m    = lane & 15;

    const float* __restrict__ Sb = S + (size_t)batch * N * N;
    const int ci0 = bi0 + m, ci1 = bi0 + 16 + m;     // S columns feeding the two A tiles
    const int cj0 = bj0 + m, cj1 = bj0 + 16 + m;     // S columns feeding the two B tiles

    v8f c00 = {}, c01 = {}, c10 = {}, c11 = {};
    for (int kk = 0; kk < N; kk += 4) {
        const int k0 = kk + 2 * half;
        const float* r0 = Sb + (size_t)k0 * N;
        const float* r1 = r0 + N;
        v2f a0, a1, b0, b1;
        a0[0] = r0[ci0]; a0[1] = r1[ci0];
        a1[0] = r0[ci1]; a1[1] = r1[ci1];
        b0[0] = r0[cj0]; b0[1] = r1[cj0];
        b1[0] = r0[cj1]; b1[1] = r1[cj1];
        c00 = __builtin_amdgcn_wmma_f32_16x16x4_f32(false, a0, false, b0, (short)0, c00, false, false);
        c01 = __builtin_amdgcn_wmma_f32_16x16x4_f32(false, a0, false, b1, (short)0, c01, false, false);
        c10 = __builtin_amdgcn_wmma_f32_16x16x4_f32(false, a1, false, b0, (short)0, c10, false, false);
        c11 = __builtin_amdgcn_wmma_f32_16x16x4_f32(false, a1, false, b1, (short)0, c11, false, false);
    }

    float* __restrict__ Out = Sigma + (size_t)batch * N * N;
#pragma unroll
    for (int r = 0; r < 8; ++r) {
        const int ra = bi0 + r + 8 * half;           // rows of tiles (0x,1x)
        Out[(size_t)ra * N + bj0 + m]            = c00[r];
        Out[(size_t)ra * N + bj0 + 16 + m]       = c01[r];
        Out[(size_t)(ra + 16) * N + bj0 + m]     = c10[r];
        Out[(size_t)(ra + 16) * N + bj0 + 16 + m] = c11[r];
    }
}

// ---------------------------------------------------------------------------
// Block-wide reductions (256 threads).
// ---------------------------------------------------------------------------
__device__ __forceinline__ float block_sum(float v, float* red, int tid) {
    red[tid] = v; __syncthreads();
    for (int s = 128; s > 0; s >>= 1) {
        if (tid < s) red[tid] += red[tid + s];
        __syncthreads();
    }
    float r = red[0]; __syncthreads();
    return r;
}
__device__ __forceinline__ float block_min(float v, float* red, int tid) {
    red[tid] = v; __syncthreads();
    for (int s = 128; s > 0; s >>= 1) {
        if (tid < s) red[tid] = fminf(red[tid], red[tid + s]);
        __syncthreads();
    }
    float r = red[0]; __syncthreads();
    return r;
}

// ---------------------------------------------------------------------------
// Kernel 2: init  w0 = b/sum(b);  nu0 = sum(b) - w0^T Sigma w0
// ---------------------------------------------------------------------------
__global__ void __launch_bounds__(256)
newton_init_kernel(const float* __restrict__ bvec, const float* __restrict__ Sigma,
                   float* __restrict__ wg, float* __restrict__ nug) {
    const int batch = blockIdx.x;
    const int tid = threadIdx.x;
    __shared__ float sw[N];
    __shared__ float red[256];

    const float bi = bvec[(size_t)batch * N + tid];
    const float sumb = block_sum(bi, red, tid);
    const float w0 = bi / sumb;
    sw[tid] = w0;
    __syncthreads();

    const float* __restrict__ Srow = Sigma + (size_t)batch * N * N + (size_t)tid * N;
    float acc = 0.f;
#pragma unroll 8
    for (int j = 0; j < N; ++j) acc = fmaf(Srow[j], sw[j], acc);
    const float q = block_sum(acc * w0, red, tid);

    wg[(size_t)batch * N + tid] = w0;
    if (tid == 0) nug[batch] = sumb - q;
}

// ---------------------------------------------------------------------------
// Kernel 3: one damped Newton step.  One 256-thread workgroup per batch;
// H (256x257 f32 ~263KB) lives in CDNA5's 320KB LDS, staged by the TDM.
// Blocked Cholesky (NB=16): wave-0 micro-factor + row-parallel panel solve +
// WMMA trailing update distributed over 8 waves.  Blocked 2-RHS solves.
// KKT elimination: x1=H^-1 r1, x2=H^-1 1, dnu=(r2-sum x1)/sum x2, dw=-x1-dnu*x2.
// LDS floats: H[N*LDA] | x1[N] | x2[N] | red[256] | ws[N]
// ---------------------------------------------------------------------------
__global__ void __launch_bounds__(256)
newton_iter_kernel(const float* __restrict__ Sigma, const float* __restrict__ bvec,
                   float* __restrict__ wg, float* __restrict__ nug) {
    extern __shared__ float smem[];
    float* Hs  = smem;
    float* x1  = Hs + N * LDA;
    float* x2  = x1 + N;
    float* red = x2 + N;
    float* ws  = red + 256;

    const int batch = blockIdx.x;
    const int tid  = threadIdx.x;
    const int wave = tid >> 5;
    const int lane = tid & 31;
    const int half = lane >> 4;
    const int m    = lane & 15;

    const float wi = wg[(size_t)batch * N + tid];
    const float bi = bvec[(size_t)batch * N + tid];
    const float nu = nug[batch];
    ws[tid] = wi;

    // --- Stage H = Sigma via one TDM DMA (HW applies the stride-257 padding) ---
    if (wave == 0) {
        tdm_load_sigma_padded(Sigma + (size_t)batch * N * N,
                              (unsigned int)(uintptr_t)Hs);
        __builtin_amdgcn_s_wait_tensorcnt(0);
    }
    __syncthreads();

    const float dget = bi / (wi * wi);
    Hs[tid * LDA + tid] += dget;
    __syncthreads();

    // --- residuals: g_i = (H w)_i - dget*w_i ---
    float acc = 0.f;
    const float* __restrict__ Hrow = Hs + tid * LDA;
#pragma unroll 8
    for (int j = 0; j < N; ++j) acc = fmaf(Hrow[j], ws[j], acc);
    const float gi = acc - dget * wi;
    const float r1 = gi - bi / wi + nu;
    const float r2 = block_sum(wi, red, tid) - 1.0f;

    x1[tid] = r1;
    x2[tid] = 1.0f;
    __syncthreads();

    // =============== Blocked in-place Cholesky of H (lower) ===============
    for (int kb = 0; kb < N; kb += NB) {
        // Phase A: 16x16 diagonal micro-factorization by wave 0 (LDS + wave sync).
        if (wave == 0) {
            float* D = Hs + kb * LDA + kb;
            for (int t = 0; t < NB; ++t) {
                if (lane == 0) D[t * LDA + t] = sqrtf(fmaxf(D[t * LDA + t], 1e-30f));
                __builtin_amdgcn_wave_barrier();
                const float dtt = D[t * LDA + t];
                if (lane > t && lane < NB) D[lane * LDA + t] /= dtt;
                __builtin_amdgcn_wave_barrier();
                if (lane > t && lane < NB) {
                    const float lt = D[lane * LDA + t];
                    for (int j = t + 1; j <= lane; ++j)
                        D[lane * LDA + j] = fmaf(-lt, D[j * LDA + t], D[lane * LDA + j]);
                }
                __builtin_amdgcn_wave_barrier();
            }
        }
        __syncthreads();

        // Phase B: panel solve, one thread per row i > kb+15 :  P <- P * D^-T
        {
            const int i = kb + NB + tid;
            if (i < N) {
                float r[NB];
#pragma unroll
                for (int j = 0; j < NB; ++j) r[j] = Hs[i * LDA + kb + j];
#pragma unroll
                for (int j = 0; j < NB; ++j) {
                    float v = r[j];
#pragma unroll
                    for (int t = 0; t < NB; ++t)
                        if (t < j) v = fmaf(-r[t], Hs[(kb + j) * LDA + kb + t], v);
                    v /= Hs[(kb + j) * LDA + kb + j];
                    r[j] = v;
                    Hs[i * LDA + kb + j] = v;
                }
            }
        }
        __syncthreads();

        // Phase C: trailing update C(I,J) -= P_I * P_J^T via FP32 WMMA, K=16.
        {
            const int nb = (N - kb - NB) >> 4;          // remaining 16-blocks
            const int ntiles = nb * (nb + 1) / 2;       // lower-triangular tiles
            for (int tix = wave; tix < ntiles; tix += 8) {
                int I = 0, base = 0;
                while (tix >= base + I + 1) { base += I + 1; ++I; }
                const int J = tix - base;
                const int ri = kb + NB + I * 16;
                const int rj = kb + NB + J * 16;
                v8f a8 = {};
#pragma unroll
                for (int kk = 0; kk < NB; kk += 4) {
                    const int k0 = kb + kk + 2 * half;
                    v2f a, bv;
                    a[0]  = Hs[(ri + m) * LDA + k0];
                    a[1]  = Hs[(ri + m) * LDA + k0 + 1];
                    bv[0] = Hs[(rj + m) * LDA + k0];
                    bv[1] = Hs[(rj + m) * LDA + k0 + 1];
                    a8 = __builtin_amdgcn_wmma_f32_16x16x4_f32(false, a, false, bv,
                                                               (short)0, a8, false, false);
                }
#pragma unroll
                for (int r = 0; r < 8; ++r)
                    Hs[(ri + r + 8 * half) * LDA + rj + m] -= a8[r];
            }
        }
        __syncthreads();
    }

    // =============== Blocked forward solve  L y = rhs  (2 RHS) ===============
    for (int kb = 0; kb < N; kb += NB) {
        if (wave == 0) {
            for (int t = 0; t < NB; ++t) {
                if (lane == 0) {
                    const float inv = 1.0f / Hs[(kb + t) * LDA + kb + t];
                    x1[kb + t] *= inv; x2[kb + t] *= inv;
                }
                __builtin_amdgcn_wave_barrier();
                if (lane > t && lane < NB) {
                    const float l = Hs[(kb + lane) * LDA + kb + t];
                    x1[kb + lane] = fmaf(-l, x1[kb + t], x1[kb + lane]);
                    x2[kb + lane] = fmaf(-l, x2[kb + t], x2[kb + lane]);
                }
                __builtin_amdgcn_wave_barrier();
            }
        }
        __syncthreads();
        const int i = kb + NB + tid;
        if (i < N) {
            float s1 = x1[i], s2 = x2[i];
#pragma unroll
            for (int t = 0; t < NB; ++t) {
                const float l = Hs[i * LDA + kb + t];
                s1 = fmaf(-l, x1[kb + t], s1);
                s2 = fmaf(-l, x2[kb + t], s2);
            }
            x1[i] = s1; x2[i] = s2;
        }
        __syncthreads();
    }

    // =============== Blocked back solve  L^T x = y  (2 RHS) ===============
    for (int kb = N - NB; kb >= 0; kb -= NB) {
        if (wave == 0) {
            for (int t = NB - 1; t >= 0; --t) {
                if (lane == 0) {
                    const float inv = 1.0f / Hs[(kb + t) * LDA + kb + t];
                    x1[kb + t] *= inv; x2[kb + t] *= inv;
                }
                __builtin_amdgcn_wave_barrier();
                if (lane < t) {
                    const float l = Hs[(kb + t) * LDA + kb + lane];  // (L^T)[lane][t]
                    x1[kb + lane] = fmaf(-l, x1[kb + t], x1[kb + lane]);
                    x2[kb + lane] = fmaf(-l, x2[kb + t], x2[kb + lane]);
                }
                __builtin_amdgcn_wave_barrier();
            }
        }
        __syncthreads();
        if (tid < kb) {
            float s1 = x1[tid], s2 = x2[tid];
#pragma unroll
            for (int t = 0; t < NB; ++t) {
                const float l = Hs[(kb + t) * LDA + tid];            // (L^T)[tid][kb+t]
                s1 = fmaf(-l, x1[kb + t], s1);
                s2 = fmaf(-l, x2[kb + t], s2);
            }
            x1[tid] = s1; x2[tid] = s2;
        }
        __syncthreads();
    }

    // =============== Newton direction, step length, update ===============
    const float s1 = block_sum(x1[tid], red, tid);
    const float s2 = block_sum(x2[tid], red, tid);
    const float dnu = (r2 - s1) / s2;
    const float dw  = -x1[tid] - dnu * x2[tid];

    const float ratio = (dw < 0.f) ? (-wi / dw) : INFINITY;
    const float alpha = fminf(1.0f, 0.9f * block_min(ratio, red, tid));

    wg[(size_t)batch * N + tid] = wi + alpha * dw;
    if (tid == 0) nug[batch] = nu + alpha * dnu;
}

// ---------------------------------------------------------------------------
// Host launcher.  d_ws layout: Sigma (256MB) | w (1MB) | nu (4KB)
// ---------------------------------------------------------------------------
extern "C" void kernel_launch(void* const* d_in, const int* in_sizes, int n_in,
                              void* d_out, int out_size, void* d_ws, size_t ws_size,
                              hipStream_t stream) {
    (void)in_sizes; (void)n_in; (void)out_size; (void)ws_size;
    const float* S    = (const float*)d_in[0];   // covmat_sqrt [B][N][N]
    const float* bvec = (const float*)d_in[1];   // b           [B][N]
    float* out   = (float*)d_out;                // w           [B][N]

    float* Sigma = (float*)d_ws;
    float* wbuf  = Sigma + (size_t)BATCH * N * N;
    float* nubuf = wbuf + (size_t)BATCH * N;

    const size_t ldsBytes = (size_t)(N * LDA + 2 * N + 256 + N) * sizeof(float); // ~267 KB
    (void)hipFuncSetAttribute((const void*)newton_iter_kernel,
                              hipFuncAttributeMaxDynamicSharedMemorySize,
                              (int)ldsBytes);

    // 1) Gram matrices via FP32 WMMA: 8 blocks/batch, each wave owns a 32x32 region.
    gram_wmma_kernel<<<BATCH * 8, 256, 0, stream>>>(S, Sigma);

    // 2) Newton init.
    newton_init_kernel<<<BATCH, 256, 0, stream>>>(bvec, Sigma, wbuf, nubuf);

    // 3) 50 damped Newton steps: TDM-staged H, WMMA-blocked Cholesky in 320KB LDS.
    for (int it = 0; it < NEWTON_ITERS; ++it)
        newton_iter_kernel<<<BATCH, 256, ldsBytes, stream>>>(Sigma, bvec, wbuf, nubuf);

    // 4) Emit final weights.
    hipMemcpyAsync(out, wbuf, (size_t)BATCH * N * sizeof(float),
                   hipMemcpyDeviceToDevice, stream);
}